// SolveDarcyFlow_17162689315495
// MI455X (gfx1250) — compile-verified
//
#include <hip/hip_runtime.h>

typedef float v2f __attribute__((ext_vector_type(2)));
typedef float v8f __attribute__((ext_vector_type(8)));

#define MM   64
#define TPB  128
#define HINV 63.0f
#define MAXIT 1600

__device__ __forceinline__ float harm(float a, float b) {
    float den = a + b;
    return den > 0.0f ? (2.0f * a * b) / den : 0.0f;
}

// D padded to 64x64: D[e][x] = forward difference / h, row 63 = 0
__device__ __forceinline__ float dpv(int e, int x) {
    if (e >= 63) return 0.0f;
    if (x == e + 1) return HINV;
    if (x == e)     return -HINV;
    return 0.0f;
}

__device__ __forceinline__ v8f wmma_f32(v2f a, v2f b, v8f c) {
    return __builtin_amdgcn_wmma_f32_16x16x4_f32(
        false, a, false, b, (short)0, c, false, false);
}

// Deterministic block reduction: wave32 xor-tree, then all threads sum the
// 4 per-wave results in fixed order. One barrier per reduction; callers
// rotate `slot` so a slot is never reused without an intervening barrier.
__device__ __forceinline__ float block_reduce(float v, float* red, int itu, int lane, int slot) {
    #pragma unroll
    for (int off = 16; off > 0; off >>= 1)
        v += __shfl_xor(v, off, 32);
    if (lane == 0) red[slot * 4 + itu] = v;
    __syncthreads();
    return red[slot * 4 + 0] + red[slot * 4 + 1] + red[slot * 4 + 2] + red[slot * 4 + 3];
}

__global__ __launch_bounds__(TPB) void darcy_pcg_kernel(
    const float* __restrict__ A, const float* __restrict__ F,
    float* __restrict__ U)
{
    __shared__ float P[MM][MM];   // search direction p
    __shared__ float S[MM][MM];   // flux scratch
    __shared__ float Q[MM][MM];   // q = L p
    __shared__ float red[12];

    const int n = blockIdx.x;
    const float* a = A + n * MM * MM;
    const float* f = F + n * MM * MM;
    float* u = U + n * MM * MM;

    const int tid  = threadIdx.x;
    const int lane = tid & 31;
    const int ml   = lane & 15;    // M (or N) index within tile
    const int hf   = lane >> 4;    // half-wave select
    // wave id as an SGPR so K-window guards lower to s_cbranch (EXEC stays all-1s
    // around WMMA; WMMA is not EXEC-skipped and requires EXEC==~0).
    const int itu  = __builtin_amdgcn_readfirstlane(tid >> 5);

    float* Pf = &P[0][0];
    float* Qf = &Q[0][0];

    // ---- preload edge weights in WMMA-output ownership pattern ----
    float axw[4][8], ayw[4][8];
    #pragma unroll
    for (int jt = 0; jt < 4; ++jt) {
        #pragma unroll
        for (int v = 0; v < 8; ++v) {
            int row = 16 * itu + 8 * hf + v;
            int col = 16 * jt + ml;
            axw[jt][v] = (col < 63) ? harm(a[row * MM + col], a[row * MM + col + 1]) : 0.0f;
            ayw[jt][v] = (row < 63) ? harm(a[row * MM + col], a[(row + 1) * MM + col]) : 0.0f;
        }
    }

    // ---- flat-ownership CG state in registers ----
    float xr[32], rr[32], dinv[32];
    float rzp = 0.0f;
    #pragma unroll
    for (int i = 0; i < 32; ++i) {
        int idx = tid + TPB * i;
        int y = idx >> 6, x = idx & 63;
        bool interior = (y > 0) && (y < 63) && (x > 0) && (x < 63);
        float d = 0.0f;
        if (interior) {
            float wl = harm(a[y * MM + x - 1], a[y * MM + x]);
            float wr = harm(a[y * MM + x], a[y * MM + x + 1]);
            float wd = harm(a[(y - 1) * MM + x], a[y * MM + x]);
            float wu = harm(a[y * MM + x], a[(y + 1) * MM + x]);
            d = 1.0f / (HINV * HINV * (wl + wr + wd + wu));
        }
        dinv[i] = d;
        xr[i] = 0.0f;
        float rv = interior ? f[idx] : 0.0f;
        rr[i] = rv;
        float z = rv * d;
        Pf[idx] = z;           // p0 = M^-1 r0
        rzp += rv * z;
    }
    __syncthreads();           // P ready; red buffer ready

    float rz = block_reduce(rzp, red, itu, lane, 2);
    const float tol = rz * 1e-12f;
    __syncthreads();

    for (int iter = 0; iter < MAXIT; ++iter) {
        // ======== Q = L * P via fp32 WMMA (banded K-windows only) ========
        // Stage 1: S = axw ⊙ (P @ D^T).  B[k][n]=D[n][k] nonzero => k in [16jt, 16jt+16]
        #pragma unroll
        for (int jt = 0; jt < 4; ++jt) {
            v8f acc = {0.f,0.f,0.f,0.f,0.f,0.f,0.f,0.f};
            #pragma unroll
            for (int w = 0; w < 5; ++w) {
                int kb = 4 * jt + w;               // compile-time per jt
                if (kb <= 15) {
                    int k0 = 4 * kb + 2 * hf;
                    v2f av, bv;
                    av.x = P[16 * itu + ml][k0];
                    av.y = P[16 * itu + ml][k0 + 1];
                    bv.x = dpv(16 * jt + ml, k0);
                    bv.y = dpv(16 * jt + ml, k0 + 1);
                    acc = wmma_f32(av, bv, acc);
                }
            }
            #pragma unroll
            for (int v = 0; v < 8; ++v)
                S[16 * itu + 8 * hf + v][16 * jt + ml] = acc[v] * axw[jt][v];
        }
        __syncthreads();

        // Stage 2: Q = S @ D.  B[k][n]=D[k][n] nonzero => k in [16jt-1, 16jt+15]
        #pragma unroll
        for (int jt = 0; jt < 4; ++jt) {
            v8f acc = {0.f,0.f,0.f,0.f,0.f,0.f,0.f,0.f};
            #pragma unroll
            for (int w = 0; w < 5; ++w) {
                int kb = 4 * jt - 1 + w;           // compile-time per jt
                if (kb >= 0 && kb <= 15) {
                    int k0 = 4 * kb + 2 * hf;
                    v2f av, bv;
                    av.x = S[16 * itu + ml][k0];
                    av.y = S[16 * itu + ml][k0 + 1];
                    bv.x = dpv(k0, 16 * jt + ml);
                    bv.y = dpv(k0 + 1, 16 * jt + ml);
                    acc = wmma_f32(av, bv, acc);
                }
            }
            #pragma unroll
            for (int v = 0; v < 8; ++v)
                Q[16 * itu + 8 * hf + v][16 * jt + ml] = acc[v];
        }
        __syncthreads();

        // Stage 3: S = ayw ⊙ (D @ P).  A[m][k]=D[16it+ml][k] nonzero => k in [16it, 16it+16]
        #pragma unroll
        for (int jt = 0; jt < 4; ++jt) {
            v8f acc = {0.f,0.f,0.f,0.f,0.f,0.f,0.f,0.f};
            #pragma unroll
            for (int w = 0; w < 5; ++w) {
                int kb = 4 * itu + w;              // uniform (SGPR) -> s_cbranch guard
                if (kb <= 15) {
                    int k0 = 4 * kb + 2 * hf;
                    v2f av, bv;
                    av.x = dpv(16 * itu + ml, k0);
                    av.y = dpv(16 * itu + ml, k0 + 1);
                    bv.x = P[k0][16 * jt + ml];
                    bv.y = P[k0 + 1][16 * jt + ml];
                    acc = wmma_f32(av, bv, acc);
                }
            }
            #pragma unroll
            for (int v = 0; v < 8; ++v)
                S[16 * itu + 8 * hf + v][16 * jt + ml] = acc[v] * ayw[jt][v];
        }
        __syncthreads();

        // Stage 4: Q += D^T @ S, then Dirichlet mask.  A[m][k]=D[k][16it+ml] => k in [16it-1, 16it+15]
        #pragma unroll
        for (int jt = 0; jt < 4; ++jt) {
            v8f acc = {0.f,0.f,0.f,0.f,0.f,0.f,0.f,0.f};
            #pragma unroll
            for (int w = 0; w < 5; ++w) {
                int kb = 4 * itu - 1 + w;          // uniform (SGPR) -> s_cbranch guard
                if (kb >= 0 && kb <= 15) {
                    int k0 = 4 * kb + 2 * hf;
                    v2f av, bv;
                    av.x = dpv(k0, 16 * itu + ml);
                    av.y = dpv(k0 + 1, 16 * itu + ml);
                    bv.x = S[k0][16 * jt + ml];
                    bv.y = S[k0 + 1][16 * jt + ml];
                    acc = wmma_f32(av, bv, acc);
                }
            }
            #pragma unroll
            for (int v = 0; v < 8; ++v) {
                int row = 16 * itu + 8 * hf + v;
                int col = 16 * jt + ml;
                float q = Q[row][col] + acc[v];
                if (row == 0 || row == 63 || col == 0 || col == 63) q = 0.0f;
                Q[row][col] = q;
            }
        }
        __syncthreads();

        // ======== CG scalar updates ========
        float pqp = 0.0f;
        #pragma unroll
        for (int i = 0; i < 32; ++i) {
            int idx = tid + TPB * i;
            pqp += Pf[idx] * Qf[idx];
        }
        float pq = block_reduce(pqp, red, itu, lane, 0);
        if (pq == 0.0f) break;                 // uniform (shared scalar)
        float alpha = rz / pq;

        float rznp = 0.0f;
        #pragma unroll
        for (int i = 0; i < 32; ++i) {
            int idx = tid + TPB * i;
            xr[i] += alpha * Pf[idx];
            float rv = rr[i] - alpha * Qf[idx];
            rr[i] = rv;
            rznp += rv * rv * dinv[i];
        }
        float rzn = block_reduce(rznp, red, itu, lane, 1);
        float beta = rzn / rz;
        rz = rzn;

        #pragma unroll
        for (int i = 0; i < 32; ++i) {
            int idx = tid + TPB * i;
            Pf[idx] = rr[i] * dinv[i] + beta * Pf[idx];
        }
        __syncthreads();

        if (rzn <= tol) break;                 // uniform (shared scalar)
    }

    #pragma unroll
    for (int i = 0; i < 32; ++i)
        u[tid + TPB * i] = xr[i];
}

extern "C" void kernel_launch(void* const* d_in, const int* in_sizes, int n_in,
                              void* d_out, int out_size, void* d_ws, size_t ws_size,
                              hipStream_t stream) {
    const float* a = (const float*)d_in[0];
    const float* f = (const float*)d_in[1];
    float* u = (float*)d_out;
    int nbatch = in_sizes[0] / (MM * MM);
    darcy_pcg_kernel<<<dim3(nbatch), dim3(TPB), 0, stream>>>(a, f, u);
}